// SimpleMRConv_9208409883078
// MI455X (gfx1250) — compile-verified
//
#include <hip/hip_runtime.h>
#include <hip/hip_bf16.h>

typedef float v2f __attribute__((ext_vector_type(2)));
typedef float v8f __attribute__((ext_vector_type(8)));

#define D 64
#define LAYERS 3

// ---------------------------------------------------------------------------
// agg = 0  (u32 zero == 0.0f, which also implements the max(agg, 0) clamp and
// the "no incoming edges" case of the reference)
// ---------------------------------------------------------------------------
__global__ void zero_kernel(float4* __restrict__ p, int n) {
    int i = blockIdx.x * blockDim.x + threadIdx.x;
    if (i < n) p[i] = make_float4(0.f, 0.f, 0.f, 0.f);
}

// ---------------------------------------------------------------------------
// Edge phase: diff = x[src] - x[dst]; agg[dst] = max(agg[dst], diff).
// Only positive diffs can survive the final max(.,0) clamp, and positive f32
// bit patterns are monotonic in u32 -> GLOBAL_ATOMIC_MAX_U32 (deterministic,
// order-independent). 16 threads per edge, float4 per thread.
// ---------------------------------------------------------------------------
__global__ void edge_max_kernel(const float* __restrict__ x,
                                const int* __restrict__ src,
                                const int* __restrict__ dst,
                                unsigned int* __restrict__ agg_u, int E) {
    int gid = blockIdx.x * blockDim.x + threadIdx.x;
    int e = gid >> 4;
    if (e >= E) return;
    int f = (gid & 15) * 4;
    int s = src[e];
    int d = dst[e];
    float4 xs = *reinterpret_cast<const float4*>(x + (size_t)s * D + f);
    float4 xd = *reinterpret_cast<const float4*>(x + (size_t)d * D + f);
    unsigned int* a = agg_u + (size_t)d * D + f;
    float d0 = xs.x - xd.x;
    float d1 = xs.y - xd.y;
    float d2 = xs.z - xd.z;
    float d3 = xs.w - xd.w;
    if (d0 > 0.f) atomicMax(a + 0, __float_as_uint(d0));
    if (d1 > 0.f) atomicMax(a + 1, __float_as_uint(d1));
    if (d2 > 0.f) atomicMax(a + 2, __float_as_uint(d2));
    if (d3 > 0.f) atomicMax(a + 3, __float_as_uint(d3));
}

// ---------------------------------------------------------------------------
// h[row, :] = concat(x[row, :], agg[row, :]) @ W(128x64) + bias
// fp32 WMMA, 16x16x4. One wave owns one 16-wide column panel of W in regs
// (32 x v2f = 64 VGPRs), loops over TILES_PER_WAVE row tiles streaming A.
// A-frag layout (16x4 f32): lane m=lane&15, khalf=lane>>4:
//   a = { A[m][4kk+2*khalf], A[m][4kk+2*khalf+1] }
// B-frag (4x16): lane n=lane&15: b = { W[4kk+2*khalf][n], W[4kk+2*khalf+1][n] }
// C/D (16x16 f32): lane col n=lane&15; VGPR v -> row v + 8*(lane>>4).
// ---------------------------------------------------------------------------
#define TILES_PER_WAVE 8

__global__ void gemm_wmma_kernel(const float* __restrict__ x,
                                 const float* __restrict__ agg,
                                 const float* __restrict__ W,     // [128][64]
                                 const float* __restrict__ bias,  // [64]
                                 float* __restrict__ h,
                                 int numRowTiles, int numChunks) {
    int wave  = (blockIdx.x * blockDim.x + threadIdx.x) >> 5;
    int lane  = threadIdx.x & 31;
    int ntile = wave & 3;          // which 16-column block of the 64 outputs
    int chunk = wave >> 2;         // which group of row tiles
    if (chunk >= numChunks) return;
    int lo = lane & 15;
    int hi = lane >> 4;
    int n  = ntile * 16 + lo;

    // Load B panel once per wave.
    v2f bfrag[32];
#pragma unroll
    for (int kk = 0; kk < 32; ++kk) {
        int kb = kk * 4 + hi * 2;
        bfrag[kk].x = W[kb * D + n];
        bfrag[kk].y = W[(kb + 1) * D + n];
    }
    float bv = bias[n];

    int rt0 = chunk * TILES_PER_WAVE;
    for (int t = 0; t < TILES_PER_WAVE; ++t) {
        int rt = rt0 + t;
        if (rt >= numRowTiles) return;
        int row = rt * 16 + lo;
        const float* xr = x   + (size_t)row * D;
        const float* ar = agg + (size_t)row * D;

        v8f c = {bv, bv, bv, bv, bv, bv, bv, bv};
#pragma unroll
        for (int kk = 0; kk < 16; ++kk) {         // K = 0..63 from x
            v2f a = *reinterpret_cast<const v2f*>(xr + kk * 4 + hi * 2);
            c = __builtin_amdgcn_wmma_f32_16x16x4_f32(
                    false, a, false, bfrag[kk], (short)0, c, false, false);
        }
#pragma unroll
        for (int kk = 16; kk < 32; ++kk) {        // K = 64..127 from agg
            v2f a = *reinterpret_cast<const v2f*>(ar + (kk - 16) * 4 + hi * 2);
            c = __builtin_amdgcn_wmma_f32_16x16x4_f32(
                    false, a, false, bfrag[kk], (short)0, c, false, false);
        }
        // Store C tile: row = rt*16 + v + 8*hi, col = n
        float* hrow = h + (size_t)(rt * 16 + hi * 8) * D + n;
#pragma unroll
        for (int v = 0; v < 8; ++v) hrow[(size_t)v * D] = c[v];
    }
}

// ---------------------------------------------------------------------------
// In-place LayerNorm + ELU, one row (64 floats) per wave, 2 elems per lane.
// ---------------------------------------------------------------------------
__global__ void ln_elu_kernel(float* __restrict__ h,
                              const float* __restrict__ gamma,
                              const float* __restrict__ beta, int Nrows) {
    int wave = (blockIdx.x * blockDim.x + threadIdx.x) >> 5;
    int lane = threadIdx.x & 31;
    if (wave >= Nrows) return;
    float* row = h + (size_t)wave * D;
    float2 v = *reinterpret_cast<const float2*>(row + lane * 2);
    float s1 = v.x + v.y;
    float s2 = v.x * v.x + v.y * v.y;
#pragma unroll
    for (int off = 16; off > 0; off >>= 1) {
        s1 += __shfl_xor(s1, off, 32);
        s2 += __shfl_xor(s2, off, 32);
    }
    float mu  = s1 * (1.f / 64.f);
    float var = s2 * (1.f / 64.f) - mu * mu;
    float inv = rsqrtf(var + 1e-5f);
    float g0 = gamma[lane * 2], g1 = gamma[lane * 2 + 1];
    float b0 = beta[lane * 2],  b1 = beta[lane * 2 + 1];
    float y0 = (v.x - mu) * inv * g0 + b0;
    float y1 = (v.y - mu) * inv * g1 + b1;
    y0 = (y0 > 0.f) ? y0 : expm1f(y0);   // ELU, alpha = 1
    y1 = (y1 > 0.f) ? y1 : expm1f(y1);
    *reinterpret_cast<float2*>(row + lane * 2) = make_float2(y0, y1);
}

// ---------------------------------------------------------------------------
extern "C" void kernel_launch(void* const* d_in, const int* in_sizes, int n_in,
                              void* d_out, int out_size, void* d_ws, size_t ws_size,
                              hipStream_t stream) {
    const float* features = (const float*)d_in[0];
    const int*   src      = (const int*)d_in[1];
    const int*   dst      = (const int*)d_in[2];
    const float* Ws       = (const float*)d_in[3];
    const float* bs       = (const float*)d_in[4];
    const float* gammas   = (const float*)d_in[5];
    const float* betas    = (const float*)d_in[6];
    float*       out      = (float*)d_out;

    const int N = in_sizes[0] / D;
    const int E = in_sizes[1];

    float* bufA = (float*)d_ws;
    float* bufB = bufA + (size_t)N * D;
    float* agg  = bufB + (size_t)N * D;

    const int numRowTiles = (N + 15) / 16;
    const int numChunks   = (numRowTiles + TILES_PER_WAVE - 1) / TILES_PER_WAVE;
    const int gemmWaves   = numChunks * 4;
    const int gemmBlocks  = (gemmWaves + 7) / 8;     // 256 threads = 8 waves

    const int zeroN      = N * (D / 4);
    const int zeroBlocks = (zeroN + 255) / 256;

    long long edgeThreads = (long long)E * 16;
    const int edgeBlocks  = (int)((edgeThreads + 255) / 256);

    const int lnBlocks = (N + 7) / 8;                 // 8 rows per 256-thr block

    const float* xin = features;
    for (int l = 0; l < LAYERS; ++l) {
        float* hout = (l == 0) ? bufA : (l == 1) ? bufB : out;

        zero_kernel<<<zeroBlocks, 256, 0, stream>>>((float4*)agg, zeroN);
        edge_max_kernel<<<edgeBlocks, 256, 0, stream>>>(
            xin, src, dst, (unsigned int*)agg, E);
        gemm_wmma_kernel<<<gemmBlocks, 256, 0, stream>>>(
            xin, agg, Ws + (size_t)l * 2 * D * D, bs + (size_t)l * D,
            hout, numRowTiles, numChunks);
        ln_elu_kernel<<<lnBlocks, 256, 0, stream>>>(
            hout, gammas + (size_t)l * D, betas + (size_t)l * D, N);

        xin = hout;
    }
}